// MyConvNet_20143396618744
// MI455X (gfx1250) — compile-verified
//
#include <hip/hip_runtime.h>

typedef __attribute__((ext_vector_type(2))) float    v2f;
typedef __attribute__((ext_vector_type(4))) float    v4f;
typedef __attribute__((ext_vector_type(8))) float    v8f;
typedef __attribute__((ext_vector_type(2))) int      v2i;
typedef __attribute__((ext_vector_type(4))) int      v4i;
typedef __attribute__((ext_vector_type(8))) int      v8i;
typedef __attribute__((ext_vector_type(4))) unsigned v4u;

#define HAVE_TDM (__has_builtin(__builtin_amdgcn_tensor_load_to_lds) && \
                  __has_builtin(__builtin_amdgcn_s_wait_tensorcnt))

// ---------------------------------------------------------------------------
// Kernel 1: conv1 (1->64, 3x3, VALID) + maxpool(3,2) + relu
//   x:[2048,1,28,28] w1:[64,1,3,3] b1:[64] -> h1:[2048,64,12,12]
// ---------------------------------------------------------------------------
__global__ __launch_bounds__(256) void k_conv1_pool(const float* __restrict__ x,
                                                    const float* __restrict__ w1,
                                                    const float* __restrict__ b1,
                                                    float* __restrict__ h1) {
  int tid = blockIdx.x * 256 + threadIdx.x;       // exactly 2048*64*144 threads
  int j = tid % 12;
  int i = (tid / 12) % 12;
  int c = (tid / 144) % 64;
  int b = tid / 9216;

  float w[9];
#pragma unroll
  for (int t = 0; t < 9; ++t) w[t] = w1[c * 9 + t];
  float bias = b1[c];

  const float* xb = x + (size_t)b * 784 + (2 * i) * 28 + 2 * j;
  float p[5][5];
#pragma unroll
  for (int r = 0; r < 5; ++r)
#pragma unroll
    for (int s = 0; s < 5; ++s) p[r][s] = xb[r * 28 + s];

  float m = -3.4e38f;
#pragma unroll
  for (int pu = 0; pu < 3; ++pu)
#pragma unroll
    for (int pv = 0; pv < 3; ++pv) {
      float acc = bias;
#pragma unroll
      for (int ky = 0; ky < 3; ++ky)
#pragma unroll
        for (int kx = 0; kx < 3; ++kx)
          acc = fmaf(p[pu + ky][pv + kx], w[ky * 3 + kx], acc);
      m = fmaxf(m, acc);
    }
  h1[tid] = fmaxf(m, 0.0f);
}

// ---------------------------------------------------------------------------
// Kernel 2: conv2 (64->64, 3x3) as implicit GEMM, V_WMMA_F32_16X16X4_F32.
//   GEMM: M=64 (oc), N=2048*100, K=576.  Weights (144 KB) staged in LDS by
//   the Tensor Data Mover (one 2D D# descriptor, ISA ch.8), waited with
//   s_wait_tensorcnt.  A fragments: ds_load_b64 (k,k+1 pairs, lane-half skew
//   folded into base).  B fragments: per-lane SADDR-form global gathers with
//   byte-offset table ds_load_b64.
// ---------------------------------------------------------------------------
__global__ __launch_bounds__(256) void k_conv2_wmma(const float* __restrict__ h1,
                                                    const float* __restrict__ w2,
                                                    const float* __restrict__ b2,
                                                    float* __restrict__ y2) {
  __shared__ int   koff[576];       // k=(ic,ky,kx) -> BYTE offset ic*144+ky*12+kx (scaled x4)
  __shared__ float lw2[64 * 576];   // all conv2 weights, 147456 B

#if HAVE_TDM
  if (threadIdx.x < 32) {           // wave 0 issues the TDM DMA (EXEC-independent)
    unsigned long long ga = (unsigned long long)(uintptr_t)w2;
    unsigned lds = (unsigned)(uintptr_t)&lw2[0];
    // D# group0: count=1 | lds_addr | global_addr[56:0] | type=2
    v4u g0;
    g0.x = 1u;
    g0.y = lds;
    g0.z = (unsigned)(ga & 0xFFFFFFFFu);
    g0.w = (unsigned)((ga >> 32) & 0x1FFFFFFu) | (2u << 30);
    // D# group1: wg_mask=0, data_size=2 (4B), no flags; tensor 576x64, tile 576x64,
    // dim0 stride 576 elements.
    v8i g1;
    g1[0] = 0x00020000;             // data_size=4B
    g1[1] = (int)(576u << 16);      // tensor_dim0[15:0] << 16 (barrier addr lo = 0)
    g1[2] = (int)(64u  << 16);      // tensor_dim0 hi=0 | tensor_dim1 lo in [31:16]
    g1[3] = (int)(576u << 16);      // tensor_dim1 hi=0 | tile_dim0 in [31:16]
    g1[4] = 64;                     // tile_dim1 | tile_dim2=0
    g1[5] = 576;                    // tensor_dim0_stride lo32
    g1[6] = 0;                      // stride0 hi | stride1 lo
    g1[7] = 0;
    v4i gz = {0, 0, 0, 0};
#if defined(__clang_major__) && (__clang_major__ >= 23)
    v8i gz8 = {0, 0, 0, 0, 0, 0, 0, 0};
    __builtin_amdgcn_tensor_load_to_lds(g0, g1, gz, gz, gz8, 0);
#else
    __builtin_amdgcn_tensor_load_to_lds(g0, g1, gz, gz, 0);
#endif
  }
  for (int t = threadIdx.x; t < 576; t += 256) {
    int ic = t / 9, r = t % 9;
    koff[t] = 4 * (ic * 144 + (r / 3) * 12 + (r % 3));
  }
  if (threadIdx.x < 32) __builtin_amdgcn_s_wait_tensorcnt(0);
  __syncthreads();
#else
  // Fallback staging: coalesced b128 global loads -> ds_store
  for (int t = threadIdx.x; t < (64 * 576) / 4; t += 256)
    ((v4f*)lw2)[t] = ((const v4f*)w2)[t];
  for (int t = threadIdx.x; t < 576; t += 256) {
    int ic = t / 9, r = t % 9;
    koff[t] = 4 * (ic * 144 + (r / 3) * 12 + (r % 3));
  }
  __syncthreads();
#endif

  int lane = threadIdx.x & 31;
  int wave = threadIdx.x >> 5;
  bool hi = lane >= 16;
  int n0 = (blockIdx.x * 8 + wave) * 16;          // N tile base (exact cover)
  int n  = n0 + (lane & 15);
  int b  = n / 100;
  int sp = n % 100;                               // oy*10 + ox
  int oy = sp / 10, ox = sp % 10;

  // per-lane byte base into h1 for the B gathers (32-bit, SADDR-friendly)
  unsigned nbase4 = ((unsigned)b * 9216u + (unsigned)(oy * 12 + ox)) * 4u;
  const char* h1b = (const char*)h1;

  // lane-half skew folded into table / weight-row bases once
  const int*   offp = koff + (hi ? 2 : 0);
  const float* arow = lw2 + (size_t)(lane & 15) * 576 + (hi ? 2 : 0);

  v8f acc0 = {}, acc1 = {}, acc2 = {}, acc3 = {};

  for (int k = 0; k < 576; k += 4) {
    v2i o = *(const v2i*)(offp + k);              // one ds_load_b64: two byte-offsets
    v2f bf;
    bf.x = *(const float*)(h1b + (nbase4 + (unsigned)o.x));
    bf.y = *(const float*)(h1b + (nbase4 + (unsigned)o.y));
    const float* ap = arow + k;                   // ds_load_b64 x4 (k,k+1 pair per lane)
    v2f a0 = *(const v2f*)(ap);
    v2f a1 = *(const v2f*)(ap + 16 * 576);
    v2f a2 = *(const v2f*)(ap + 32 * 576);
    v2f a3 = *(const v2f*)(ap + 48 * 576);
    acc0 = __builtin_amdgcn_wmma_f32_16x16x4_f32(false, a0, false, bf, (short)0, acc0, false, false);
    acc1 = __builtin_amdgcn_wmma_f32_16x16x4_f32(false, a1, false, bf, (short)0, acc1, false, false);
    acc2 = __builtin_amdgcn_wmma_f32_16x16x4_f32(false, a2, false, bf, (short)0, acc2, false, false);
    acc3 = __builtin_amdgcn_wmma_f32_16x16x4_f32(false, a3, false, bf, (short)0, acc3, false, false);
  }

  // C/D layout: VGPR r, lanes 0-15 -> M=r, lanes 16-31 -> M=r+8; N = lane&15
  size_t outb = (size_t)b * 6400 + sp;            // (b*64 + m)*100 + sp
#pragma unroll
  for (int r = 0; r < 8; ++r) {
    int mb = r + (hi ? 8 : 0);
    y2[outb + (size_t)(mb + 0)  * 100] = acc0[r] + b2[mb + 0];
    y2[outb + (size_t)(mb + 16) * 100] = acc1[r] + b2[mb + 16];
    y2[outb + (size_t)(mb + 32) * 100] = acc2[r] + b2[mb + 32];
    y2[outb + (size_t)(mb + 48) * 100] = acc3[r] + b2[mb + 48];
  }
}

// ---------------------------------------------------------------------------
// Kernel 3: maxpool(3,2) + relu : y2:[2048,64,10,10] -> h2:[2048,64,4,4]
// ---------------------------------------------------------------------------
__global__ __launch_bounds__(256) void k_pool2(const float* __restrict__ y2,
                                               float* __restrict__ h2) {
  int tid = blockIdx.x * 256 + threadIdx.x;       // exactly 2048*64*16 threads
  int j = tid & 3;
  int i = (tid >> 2) & 3;
  int bc = tid >> 4;
  const float* src = y2 + (size_t)bc * 100 + (2 * i) * 10 + 2 * j;
  float m = -3.4e38f;
#pragma unroll
  for (int p = 0; p < 3; ++p)
#pragma unroll
    for (int q = 0; q < 3; ++q) m = fmaxf(m, src[p * 10 + q]);
  h2[tid] = fmaxf(m, 0.0f);
}

// ---------------------------------------------------------------------------
// Kernel 4: conv3 (64x4x4 -> 10) as GEMM with V_WMMA_F32_16X16X4_F32
//   out[b][o] = b3[o] + sum_k h2[b][k]*w3[o][k],  K=1024, M=16 (10 used), N=2048
// ---------------------------------------------------------------------------
__global__ __launch_bounds__(128) void k_fc_wmma(const float* __restrict__ h2,
                                                 const float* __restrict__ w3,
                                                 const float* __restrict__ b3,
                                                 float* __restrict__ out) {
  int lane = threadIdx.x & 31;
  int wave = threadIdx.x >> 5;
  bool hi = lane >= 16;
  int n0 = (blockIdx.x * 4 + wave) * 16;
  int b  = n0 + (lane & 15);
  int mrow = lane & 15;
  const float* bp = h2 + (size_t)b * 1024;

  v8f acc = {};
  v2f zero = {0.0f, 0.0f};
  for (int k = 0; k < 1024; k += 4) {
    int ka = k + (hi ? 2 : 0);
    v2f bf = *(const v2f*)(bp + ka);
    v2f af = zero;                                 // pad rows M=10..15 with 0
    if (mrow < 10) af = *(const v2f*)(w3 + (size_t)mrow * 1024 + ka);
    acc = __builtin_amdgcn_wmma_f32_16x16x4_f32(false, af, false, bf, (short)0, acc, false, false);
  }

#pragma unroll
  for (int r = 0; r < 8; ++r) {
    int m = r + (hi ? 8 : 0);
    if (m < 10) out[(size_t)b * 10 + m] = acc[r] + b3[m];
  }
}

// ---------------------------------------------------------------------------
extern "C" void kernel_launch(void* const* d_in, const int* in_sizes, int n_in,
                              void* d_out, int out_size, void* d_ws, size_t ws_size,
                              hipStream_t stream) {
  const float* x  = (const float*)d_in[0];
  const float* w1 = (const float*)d_in[1];
  const float* b1 = (const float*)d_in[2];
  const float* w2 = (const float*)d_in[3];
  const float* b2 = (const float*)d_in[4];
  const float* w3 = (const float*)d_in[5];
  const float* b3 = (const float*)d_in[6];
  float* out = (float*)d_out;

  char* ws = (char*)d_ws;
  float* h1 = (float*)(ws);                                   // 75,497,472 B
  float* y2 = (float*)(ws + 75497472u);                       // 52,428,800 B
  float* h2 = (float*)(ws + 75497472u + 52428800u);           //  8,388,608 B

  k_conv1_pool<<<73728, 256, 0, stream>>>(x, w1, b1, h1);
  // conv2 WMMA: 12800 wave-tiles / 8 waves per block -> 1600 blocks
  k_conv2_wmma<<<1600, 256, 0, stream>>>(h1, w2, b2, y2);
  k_pool2<<<8192, 256, 0, stream>>>(y2, h2);
  k_fc_wmma<<<32, 128, 0, stream>>>(h2, w3, b3, out);
}